// GraphConvolution_47708496724381
// MI455X (gfx1250) — compile-verified
//
#include <hip/hip_runtime.h>

typedef __attribute__((ext_vector_type(2))) float v2f;
typedef __attribute__((ext_vector_type(4))) float v4f;
typedef __attribute__((ext_vector_type(8))) float v8f;

#define DIM 128  // D_IN == D_OUT == 128

// ---------------------------------------------------------------------------
// Kernel 1: support = x @ W using V_WMMA_F32_16X16X4_F32 (exact fp32 WMMA).
// One wave computes one 16x16 output tile. 8 waves per block cover the full
// N=128 width of one 16-row M stripe. K=128 -> 32 WMMA steps of K=4.
//
// f32 16x4 A layout : lanes 0-15 hold M=lane,   VGPR0=K0, VGPR1=K1
//                     lanes 16-31 hold M=lane-16, VGPR0=K2, VGPR1=K3
// f32 4x16 B layout : lanes 0-15 hold N=lane,   VGPR0=K0, VGPR1=K1
//                     lanes 16-31 hold N=lane-16, VGPR0=K2, VGPR1=K3
// f32 16x16 C/D     : lanes 0-15 N=lane (M=r), lanes 16-31 N=lane-16 (M=8+r)
// ---------------------------------------------------------------------------
__global__ __launch_bounds__(256)
void gemm_support_wmma(const float* __restrict__ x,
                       const float* __restrict__ w,
                       float* __restrict__ support,
                       int n_nodes) {
    const int wave = threadIdx.x >> 5;      // 0..7 -> N tile index
    const int lane = threadIdx.x & 31;
    const int half = lane >> 4;             // 0: K lo pair, 1: K hi pair
    const int l    = lane & 15;

    const int m0 = blockIdx.x * 16;
    const int n0 = wave * 16;

    // Clamp row for tail safety (n_nodes = 50000 is a multiple of 16, but be safe).
    int mrow = m0 + l;
    if (mrow >= n_nodes) mrow = n_nodes - 1;

    // a.x = x[mrow][half*2 + 4*kk], a.y = next K
    const float* xrow = x + (long long)mrow * DIM + half * 2;
    // b.x = W[half*2 + 4*kk][n0 + l], b.y = next K row
    const float* wcol = w + (half * 2) * DIM + n0 + l;

    v8f c = {};
#pragma unroll
    for (int kk = 0; kk < 32; ++kk) {
        v2f a, b;
        a.x = xrow[kk * 4 + 0];
        a.y = xrow[kk * 4 + 1];
        b.x = wcol[(kk * 4 + 0) * DIM];
        b.y = wcol[(kk * 4 + 1) * DIM];
        // 8 args: (neg_a, A, neg_b, B, c_mod, C, reuse_a, reuse_b)
        c = __builtin_amdgcn_wmma_f32_16x16x4_f32(
                false, a, false, b, (short)0, c, false, false);
    }

    // Store: lane l -> col n0+l ; VGPR r -> row m0 + half*8 + r
    const int rbase = m0 + half * 8;
    float* outp = support + (long long)rbase * DIM + n0 + l;
#pragma unroll
    for (int r = 0; r < 8; ++r) {
        if (rbase + r < n_nodes) outp[(long long)r * DIM] = c[r];
    }
}

// ---------------------------------------------------------------------------
// Kernel 2: out[n][d] = bias[d]  (scatter kernel accumulates on top).
// ---------------------------------------------------------------------------
__global__ __launch_bounds__(256)
void init_bias(const float* __restrict__ bias, float* __restrict__ out,
               long long total) {
    long long t = (long long)blockIdx.x * blockDim.x + threadIdx.x;
    if (t < total) out[t] = bias[t & (DIM - 1)];
}

// ---------------------------------------------------------------------------
// Kernel 3: edge scatter. 32 lanes per edge, 4 floats per lane (float4 gather
// from support — L2-resident — then 4 hardware global_atomic_add_f32).
// ---------------------------------------------------------------------------
__global__ __launch_bounds__(256)
void spmm_scatter(const float* __restrict__ support,
                  const int* __restrict__ src,
                  const int* __restrict__ dst,
                  const float* __restrict__ ev,
                  float* __restrict__ out,
                  int n_edges) {
    long long t = (long long)blockIdx.x * blockDim.x + threadIdx.x;
    int e = (int)(t >> 5);
    int q = (int)(t & 31);
    if (e >= n_edges) return;

    const int   s   = src[e];
    const int   d   = dst[e];
    const float val = ev[e];

    const v4f m = *((const v4f*)(support + (long long)s * DIM) + q);
    float* op = out + (long long)d * DIM + q * 4;

    unsafeAtomicAdd(op + 0, m.x * val);
    unsafeAtomicAdd(op + 1, m.y * val);
    unsafeAtomicAdd(op + 2, m.z * val);
    unsafeAtomicAdd(op + 3, m.w * val);
}

// ---------------------------------------------------------------------------
extern "C" void kernel_launch(void* const* d_in, const int* in_sizes, int n_in,
                              void* d_out, int out_size, void* d_ws, size_t ws_size,
                              hipStream_t stream) {
    const float* x    = (const float*)d_in[0];
    const float* w    = (const float*)d_in[1];
    const float* bias = (const float*)d_in[2];
    const int*   src  = (const int*)d_in[3];
    const int*   dst  = (const int*)d_in[4];
    const float* ev   = (const float*)d_in[5];

    float* out     = (float*)d_out;
    float* support = (float*)d_ws;   // n_nodes * 128 floats (25.6 MB)

    const int n_nodes = in_sizes[0] / DIM;
    const int n_edges = in_sizes[3];

    // 1) support = x @ W   (WMMA fp32)
    const int m_tiles = (n_nodes + 15) / 16;
    gemm_support_wmma<<<m_tiles, 256, 0, stream>>>(x, w, support, n_nodes);

    // 2) out = bias (broadcast over rows)
    const long long total = (long long)n_nodes * DIM;
    init_bias<<<(int)((total + 255) / 256), 256, 0, stream>>>(bias, out, total);

    // 3) out[dst[e]] += ev[e] * support[src[e]]
    const long long work = (long long)n_edges * 32;
    spmm_scatter<<<(int)((work + 255) / 256), 256, 0, stream>>>(
        support, src, dst, ev, out, n_edges);
}